// GraphModel_31404800868418
// MI455X (gfx1250) — compile-verified
//
#include <hip/hip_runtime.h>
#include <math.h>

#define D 64
#define CLS 2

typedef __attribute__((ext_vector_type(2))) float v2f;
typedef __attribute__((ext_vector_type(8))) float v8f;

// ---------- degree / normalization ----------

__global__ void deg_init_k(float* deg, int N) {
  int i = blockIdx.x * blockDim.x + threadIdx.x;
  if (i < N) deg[i] = 1.0f;  // self-loop contributes 1 to every node's degree
}

__global__ void deg_accum_k(const int* __restrict__ col, float* deg, int E) {
  int e = blockIdx.x * blockDim.x + threadIdx.x;
  if (e < E) atomicAdd(&deg[col[e]], 1.0f);
}

__global__ void dinv_k(float* deg, int N) {
  int i = blockIdx.x * blockDim.x + threadIdx.x;
  if (i < N) deg[i] = rsqrtf(deg[i]);  // deg >= 1 always (self-loops)
}

// ---------- propagation: y = Ahat * x ----------

// self-loop term + zero-init in one pass: y[i,f] = dinv[i]^2 * x[i,f]
__global__ void self_init_k(const float* __restrict__ xin, const float* __restrict__ dinv,
                            float* __restrict__ y, int total) {
  int i = blockIdx.x * blockDim.x + threadIdx.x;
  if (i >= total) return;
  float d = dinv[i >> 6];  // D == 64
  y[i] = d * d * xin[i];
}

// 16 lanes per edge, 4 floats per lane (float4 gather, 4 f32 atomics scatter)
__global__ void edge_scatter_k(const int* __restrict__ row, const int* __restrict__ col,
                               const float* __restrict__ dinv, const float* __restrict__ xin,
                               float* __restrict__ y, int E) {
  int idx = blockIdx.x * blockDim.x + threadIdx.x;
  int e = idx >> 4;
  if (e >= E) return;
  int t = idx & 15;
  int s = row[e];
  int d = col[e];
  float w = dinv[s] * dinv[d];
  float4 v = ((const float4*)(xin + (size_t)s * D))[t];
  float* yd = y + (size_t)d * D + (size_t)t * 4;
  atomicAdd(yd + 0, w * v.x);
  atomicAdd(yd + 1, w * v.y);
  atomicAdd(yd + 2, w * v.z);
  atomicAdd(yd + 3, w * v.w);
}

// ---------- projection (WMMA f32 16x16x4) + 2-class log_softmax ----------
// One wave per 16-node tile. M=16 nodes, K=64 (16 chained WMMAs), N=16 (classes
// padded 2->16 with zero B columns via a branchless 0/1 mask). Bias preloaded
// into the accumulator.
__global__ void proj_lsm_k(const float* __restrict__ x, const float* __restrict__ W,
                           const float* __restrict__ bias, float* __restrict__ out, int N) {
  const int lane = threadIdx.x & 31;
  const int wave = threadIdx.x >> 5;
  const int tile = blockIdx.x * (blockDim.x >> 5) + wave;
  const int row0 = tile * 16;
  if (row0 >= N) return;           // wave-uniform: EXEC stays all-ones below
  const int half = lane >> 4;      // 0: lanes 0-15, 1: lanes 16-31
  const int l = lane & 15;         // A: row M=l ; B/C/D: column n=l

  // Branchless padding: clamp W row to a valid one, zero it with a mask.
  const int wl = (l < CLS) ? l : 0;
  const float wmask = (l < CLS) ? 1.0f : 0.0f;

  const float bn = wmask * bias[wl];
  v8f c = {bn, bn, bn, bn, bn, bn, bn, bn};

  const float* __restrict__ xr = x + (size_t)(row0 + l) * D;  // A row M=l
  const float* __restrict__ wr = W + (size_t)wl * D;          // B column n=l (masked)

#pragma unroll
  for (int k0 = 0; k0 < D; k0 += 4) {
    // ISA 16x4 f32 A layout: VGPR0 holds K = k0 + 2*half, VGPR1 holds K+1
    const int kk = k0 + half * 2;
    v2f a, b;
    a.x = xr[kk + 0];
    a.y = xr[kk + 1];
    b.x = wmask * wr[kk + 0];  // B[k][n] = W[n][k], zero for padded n >= CLS
    b.y = wmask * wr[kk + 1];
    c = __builtin_amdgcn_wmma_f32_16x16x4_f32(
        /*neg_a=*/false, a, /*neg_b=*/false, b,
        /*c_mod=*/(short)0, c, /*reuse_a=*/false, /*reuse_b=*/false);
  }

  // D layout: lane(l,half) holds column n=l for rows M = half*8 + r, r=0..7.
  // 2-class log_softmax: pair-exchange logits via XOR-1 lane shuffle.
#pragma unroll
  for (int r = 0; r < 8; ++r) {
    float z = c[r];
    float other = __shfl_xor(z, 1, 32);  // lanes (0,1) and (16,17) swap class logits
    float m = fmaxf(z, other);
    float lse = m + logf(expf(z - m) + expf(other - m));
    float res = z - lse;
    if (l < CLS) {
      int node = row0 + half * 8 + r;
      if (node < N) out[(size_t)node * CLS + l] = res;
    }
  }
}

// ---------- launcher ----------

extern "C" void kernel_launch(void* const* d_in, const int* in_sizes, int n_in,
                              void* d_out, int out_size, void* d_ws, size_t ws_size,
                              hipStream_t stream) {
  const float* x   = (const float*)d_in[0];   // [N, 64] f32
  const int*   ei  = (const int*)d_in[1];     // [2, E] indices
  const float* W   = (const float*)d_in[2];   // [2, 64] f32 (torch Linear weight)
  const float* bia = (const float*)d_in[3];   // [2] f32
  float* out = (float*)d_out;                 // [N, 2] f32

  const int N = in_sizes[0] / D;
  const int E = in_sizes[1] / 2;
  const int* row = ei;       // sources
  const int* col = ei + E;   // targets

  // workspace: dinv [N] | buf1 [N*64] | buf2 [N*64]
  float* dinv = (float*)d_ws;
  size_t off = (((size_t)N * sizeof(float)) + 255) & ~(size_t)255;
  float* buf1 = (float*)((char*)d_ws + off);
  float* buf2 = buf1 + (size_t)N * D;

  const int B = 256;
  deg_init_k<<<(N + B - 1) / B, B, 0, stream>>>(dinv, N);
  deg_accum_k<<<(E + B - 1) / B, B, 0, stream>>>(col, dinv, E);
  dinv_k<<<(N + B - 1) / B, B, 0, stream>>>(dinv, N);

  const int total = N * D;
  const long ethreads = (long)E * 16;
  const int eblocks = (int)((ethreads + B - 1) / B);

  // hop 1: buf1 = Ahat * x
  self_init_k<<<(total + B - 1) / B, B, 0, stream>>>(x, dinv, buf1, total);
  edge_scatter_k<<<eblocks, B, 0, stream>>>(row, col, dinv, x, buf1, E);

  // hop 2: buf2 = Ahat * buf1
  self_init_k<<<(total + B - 1) / B, B, 0, stream>>>(buf1, dinv, buf2, total);
  edge_scatter_k<<<eblocks, B, 0, stream>>>(row, col, dinv, buf1, buf2, E);

  // projection + log_softmax (WMMA f32)
  const int tiles = (N + 15) / 16;
  const int wavesPerBlock = 8;  // 256 threads
  proj_lsm_k<<<(tiles + wavesPerBlock - 1) / wavesPerBlock, wavesPerBlock * 32, 0, stream>>>(
      buf2, W, bia, out, N);
}